// layer_GCN_34986803593393
// MI455X (gfx1250) — compile-verified
//
#include <hip/hip_runtime.h>
#include <hip/hip_bf16.h>
#include <math.h>

// Problem constants (from reference)
#define CIRC 10000
#define DIS  4000
#define LAT  64
#define EPS_F 1e-8f

typedef __attribute__((ext_vector_type(2))) float v2f;
typedef __attribute__((ext_vector_type(8))) float v8f;

// ---------------------------------------------------------------------------
// Degree kernels: inv_d[j] over columns, inv_c[i] over rows of A (10000x4000)
// ---------------------------------------------------------------------------
__global__ __launch_bounds__(256)
void col_degree_kernel(const float* __restrict__ A, float* __restrict__ inv_d) {
    int j = blockIdx.x * blockDim.x + threadIdx.x;
    if (j >= DIS) return;
    float s = 0.f;
    for (int i = 0; i < CIRC; ++i) s += A[(size_t)i * DIS + j];
    inv_d[j] = (s > 0.f) ? rsqrtf(fmaxf(s, EPS_F)) : 0.f;
}

__global__ __launch_bounds__(256)
void row_degree_kernel(const float* __restrict__ A, float* __restrict__ inv_c) {
    __shared__ float red[256];
    int row = blockIdx.x;
    float s = 0.f;
    for (int c = threadIdx.x; c < DIS; c += 256) s += A[(size_t)row * DIS + c];
    red[threadIdx.x] = s;
    __syncthreads();
    for (int off = 128; off > 0; off >>= 1) {
        if (threadIdx.x < off) red[threadIdx.x] += red[threadIdx.x + off];
        __syncthreads();
    }
    if (threadIdx.x == 0) {
        float t = red[0];
        inv_c[row] = (t > 0.f) ? rsqrtf(fmaxf(t, EPS_F)) : 0.f;
    }
}

// ---------------------------------------------------------------------------
// Init: lc = circ_emb, ld = dis_emb, acc (in d_out) = 0
// ---------------------------------------------------------------------------
__global__ __launch_bounds__(256)
void init_state_kernel(const float* __restrict__ ce, const float* __restrict__ de,
                       float* __restrict__ lc, float* __restrict__ ld,
                       float* __restrict__ accc, float* __restrict__ accd) {
    int t = blockIdx.x * 256 + threadIdx.x;
    if (t < CIRC * LAT) { lc[t] = ce[t]; accc[t] = 0.f; }
    else if (t < (CIRC + DIS) * LAT) {
        int u = t - CIRC * LAT;
        ld[u] = de[u]; accd[u] = 0.f;
    }
}

// ---------------------------------------------------------------------------
// Pre-scale operand rows: lcp[i,:] = inv_c[i]*lc[i,:]; ldp[k,:] = inv_d[k]*ld[k,:]
// ---------------------------------------------------------------------------
__global__ __launch_bounds__(256)
void prescale_kernel(const float* __restrict__ lc, const float* __restrict__ ld,
                     const float* __restrict__ inv_c, const float* __restrict__ inv_d,
                     float* __restrict__ lcp, float* __restrict__ ldp) {
    int t = blockIdx.x * 256 + threadIdx.x;
    if (t < CIRC * LAT) lcp[t] = lc[t] * inv_c[t >> 6];
    else if (t < (CIRC + DIS) * LAT) {
        int u = t - CIRC * LAT;
        ldp[u] = ld[u] * inv_d[u >> 6];
    }
}

// ---------------------------------------------------------------------------
// Fused normalized-adjacency GEMM:  C[M x 64] = rowscale .* (Aeff @ B)
//   TRANS=false: Aeff[m][k] = A[m*DIS + k]   (M=10000, K=4000)   -> new_c
//   TRANS=true : Aeff[m][k] = A[k*DIS + m]   (M=4000,  K=10000)  -> new_d
// Block = 128 threads = 4 waves; each wave owns one 16x16 N-tile; K tiled by 16
// through LDS. Uses V_WMMA_F32_16X16X4_F32 (fp32 WMMA, wave32).
// ---------------------------------------------------------------------------
template <bool TRANS>
__global__ __launch_bounds__(128)
void gemm_norm_wmma(const float* __restrict__ A, const float* __restrict__ B,
                    const float* __restrict__ rowscale, float* __restrict__ C,
                    int K) {
    __shared__ float As[16][17];   // +1 pad: bank-conflict-free transposed fills
    __shared__ float Bs[16][64];
    const int tid  = threadIdx.x;
    const int wave = tid >> 5;
    const int lane = tid & 31;
    const int lh   = lane >> 4;    // 0 | 1  (K-pair selector)
    const int ll   = lane & 15;    // M (A-frag) or N (B-frag) index
    const int M0   = blockIdx.x * 16;
    const int N0   = wave * 16;

    v8f acc = {};
    for (int k0 = 0; k0 < K; k0 += 16) {
        // Stage 16x16 A tile (transposing on the fly for A^T operand)
        for (int idx = tid; idx < 256; idx += 128) {
            if (TRANS) {
                int kk = idx >> 4, r = idx & 15;
                As[r][kk] = A[(size_t)(k0 + kk) * DIS + (M0 + r)];
            } else {
                int r = idx >> 4, kk = idx & 15;
                As[r][kk] = A[(size_t)(M0 + r) * DIS + (k0 + kk)];
            }
        }
        // Stage 16x64 B tile (pre-scaled embeddings)
        for (int idx = tid; idx < 1024; idx += 128) {
            int kk = idx >> 6, n = idx & 63;
            Bs[kk][n] = B[(size_t)(k0 + kk) * LAT + n];
        }
        __syncthreads();
#pragma unroll
        for (int ks = 0; ks < 16; ks += 4) {
            v2f a, b;
            a.x = As[ll][ks + 2 * lh];
            a.y = As[ll][ks + 2 * lh + 1];
            b.x = Bs[ks + 2 * lh][N0 + ll];
            b.y = Bs[ks + 2 * lh + 1][N0 + ll];
            acc = __builtin_amdgcn_wmma_f32_16x16x4_f32(
                false, a, false, b, (short)0, acc, false, false);
        }
        __syncthreads();
    }
    // D layout: VGPR v -> row M0 + v + 8*lh, col N0 + ll
#pragma unroll
    for (int v = 0; v < 8; ++v) {
        int row = M0 + v + 8 * lh;
        C[(size_t)row * LAT + N0 + ll] = acc[v] * rowscale[row];
    }
}

// ---------------------------------------------------------------------------
// Cosine weighting + accumulate: one wave per row (64 floats = 32 x float2)
//   w = <layer,ego> / (max(|layer|,eps)*max(|ego|,eps)); layer *= w;
//   acc += layer; next-layer input = layer
// ---------------------------------------------------------------------------
__global__ __launch_bounds__(256)
void cosine_acc_kernel(const float* __restrict__ newc, const float* __restrict__ newd,
                       const float* __restrict__ ce, const float* __restrict__ de,
                       float* __restrict__ accc, float* __restrict__ accd,
                       float* __restrict__ lc, float* __restrict__ ld) {
    const int wave = threadIdx.x >> 5;
    const int lane = threadIdx.x & 31;
    int row = blockIdx.x * 8 + wave;
    if (row >= CIRC + DIS) return;

    const float* src; const float* ego; float* acc; float* dst; int r;
    if (row < CIRC) { r = row;        src = newc; ego = ce; acc = accc; dst = lc; }
    else            { r = row - CIRC; src = newd; ego = de; acc = accd; dst = ld; }

    float2 x = ((const float2*)(src + (size_t)r * LAT))[lane];
    float2 e = ((const float2*)(ego + (size_t)r * LAT))[lane];
    float dot = x.x * e.x + x.y * e.y;
    float nx  = x.x * x.x + x.y * x.y;
    float ne  = e.x * e.x + e.y * e.y;
#pragma unroll
    for (int o = 16; o > 0; o >>= 1) {
        dot += __shfl_xor(dot, o, 32);
        nx  += __shfl_xor(nx,  o, 32);
        ne  += __shfl_xor(ne,  o, 32);
    }
    float na = fmaxf(sqrtf(nx), EPS_F);
    float nb = fmaxf(sqrtf(ne), EPS_F);
    float w  = dot / (na * nb);
    x.x *= w; x.y *= w;
    ((float2*)(dst + (size_t)r * LAT))[lane] = x;
    float2 a = ((float2*)(acc + (size_t)r * LAT))[lane];
    a.x += x.x; a.y += x.y;
    ((float2*)(acc + (size_t)r * LAT))[lane] = a;
}

// ---------------------------------------------------------------------------
// T[M x 64] = Am[M x 64] @ Bm[64 x 64]   (K=64, direct-register WMMA)
// ---------------------------------------------------------------------------
__global__ __launch_bounds__(128)
void gemm64_wmma(const float* __restrict__ Am, const float* __restrict__ Bm,
                 float* __restrict__ Cm) {
    const int tid  = threadIdx.x;
    const int wave = tid >> 5, lane = tid & 31;
    const int lh = lane >> 4, ll = lane & 15;
    const int M0 = blockIdx.x * 16, N0 = wave * 16;
    const int m = M0 + ll;
    v8f acc = {};
#pragma unroll
    for (int k0 = 0; k0 < 64; k0 += 4) {
        v2f a, b;
        const float* ap = Am + (size_t)m * LAT + k0 + 2 * lh;   // contiguous pair
        a.x = ap[0]; a.y = ap[1];
        int kb = k0 + 2 * lh;
        b.x = Bm[(size_t)kb * LAT + N0 + ll];
        b.y = Bm[(size_t)(kb + 1) * LAT + N0 + ll];
        acc = __builtin_amdgcn_wmma_f32_16x16x4_f32(
            false, a, false, b, (short)0, acc, false, false);
    }
#pragma unroll
    for (int v = 0; v < 8; ++v)
        Cm[(size_t)(M0 + v + 8 * lh) * LAT + N0 + ll] = acc[v];
}

// ---------------------------------------------------------------------------
// predict[10000 x 4000] = T[10000 x 64] @ dis_all[4000 x 64]^T
// B-transposed operand == per-lane contiguous float2 loads (no LDS needed).
// ---------------------------------------------------------------------------
__global__ __launch_bounds__(128)
void gemm_bt_wmma(const float* __restrict__ T, const float* __restrict__ D,
                  float* __restrict__ P) {
    const int tid  = threadIdx.x;
    const int wave = tid >> 5, lane = tid & 31;
    const int lh = lane >> 4, ll = lane & 15;
    const int M0 = blockIdx.x * 16;
    const int nt = blockIdx.y * 4 + wave;        // N-tile index
    if (nt >= DIS / 16) return;                  // wave-uniform guard
    const int N0 = nt * 16;
    const int m = M0 + ll, n = N0 + ll;
    v8f acc = {};
#pragma unroll
    for (int k0 = 0; k0 < 64; k0 += 4) {
        v2f a, b;
        const float* ap = T + (size_t)m * LAT + k0 + 2 * lh;
        a.x = ap[0]; a.y = ap[1];
        const float* bp = D + (size_t)n * LAT + k0 + 2 * lh;
        b.x = bp[0]; b.y = bp[1];
        acc = __builtin_amdgcn_wmma_f32_16x16x4_f32(
            false, a, false, b, (short)0, acc, false, false);
    }
#pragma unroll
    for (int v = 0; v < 8; ++v)
        P[(size_t)(M0 + v + 8 * lh) * DIS + N0 + ll] = acc[v];
}

// ---------------------------------------------------------------------------
extern "C" void kernel_launch(void* const* d_in, const int* in_sizes, int n_in,
                              void* d_out, int out_size, void* d_ws, size_t ws_size,
                              hipStream_t stream) {
    const float* A  = (const float*)d_in[0];   // 10000 x 4000
    const float* ce = (const float*)d_in[1];   // 10000 x 64
    const float* de = (const float*)d_in[2];   // 4000 x 64
    const float* rc = (const float*)d_in[3];   // 64 x 64

    float* out  = (float*)d_out;
    float* accc = out;                         // circ_all 10000x64
    float* accd = out + CIRC * LAT;            // dis_all  4000x64
    float* pred = out + (CIRC + DIS) * LAT;    // predict  10000x4000

    // Workspace layout (floats)
    float* ws    = (float*)d_ws;
    float* inv_c = ws;                 size_t o = CIRC;
    float* inv_d = ws + o;             o += DIS;
    float* lc    = ws + o;             o += CIRC * LAT;
    float* ld    = ws + o;             o += DIS * LAT;
    float* lcp   = ws + o;             o += CIRC * LAT;
    float* ldp   = ws + o;             o += DIS * LAT;
    float* newc  = ws + o;             o += CIRC * LAT;
    float* newd  = ws + o;             o += DIS * LAT;
    float* Tbuf  = ws + o;             // CIRC*LAT

    const int EMB_TOTAL = (CIRC + DIS) * LAT;  // 896000

    // 1) Degrees -> inv_c / inv_d
    col_degree_kernel<<<(DIS + 255) / 256, 256, 0, stream>>>(A, inv_d);
    row_degree_kernel<<<CIRC, 256, 0, stream>>>(A, inv_c);

    // 2) Init layer inputs + zero accumulators (d_out is poisoned)
    init_state_kernel<<<(EMB_TOTAL + 255) / 256, 256, 0, stream>>>(
        ce, de, lc, ld, accc, accd);

    // 3) Three propagation layers
    for (int layer = 0; layer < 3; ++layer) {
        prescale_kernel<<<(EMB_TOTAL + 255) / 256, 256, 0, stream>>>(
            lc, ld, inv_c, inv_d, lcp, ldp);
        // new_c = diag(inv_c) * A * (diag(inv_d)*ld)
        gemm_norm_wmma<false><<<CIRC / 16, 128, 0, stream>>>(A, ldp, inv_c, newc, DIS);
        // new_d = diag(inv_d) * A^T * (diag(inv_c)*lc)
        gemm_norm_wmma<true ><<<DIS / 16, 128, 0, stream>>>(A, lcp, inv_d, newd, CIRC);
        cosine_acc_kernel<<<(CIRC + DIS) / 8, 256, 0, stream>>>(
            newc, newd, ce, de, accc, accd, lc, ld);
    }

    // 4) predict = (circ_all @ re_CD) @ dis_all^T
    gemm64_wmma<<<CIRC / 16, 128, 0, stream>>>(accc, rc, Tbuf);
    dim3 gp(CIRC / 16, (DIS / 16 + 3) / 4);
    gemm_bt_wmma<<<gp, 128, 0, stream>>>(Tbuf, accd, pred);
}